// GCN_90778428768712
// MI455X (gfx1250) — compile-verified
//
#include <hip/hip_runtime.h>
#include <hip/hip_bf16.h>

typedef __attribute__((ext_vector_type(2))) float v2f;
typedef __attribute__((ext_vector_type(8))) float v8f;

#define N_NODES 100000
#define N_EDGES 3200000
#define IN_C 256
#define HID 32
#define OUT_C 64

// ---------------------------------------------------------------------------
// 1) degree count via f32 atomics (integer-valued adds -> order invariant)
// ---------------------------------------------------------------------------
__global__ void gcn_deg_kernel(const long long* __restrict__ dst,
                               float* __restrict__ deg) {
    int e = blockIdx.x * blockDim.x + threadIdx.x;
    if (e < N_EDGES) {
        atomicAdd(&deg[dst[e]], 1.0f);
    }
}

__global__ void gcn_dinv_kernel(float* __restrict__ deg_dinv) {
    int i = blockIdx.x * blockDim.x + threadIdx.x;
    if (i < N_NODES) {
        float d = deg_dinv[i];
        deg_dinv[i] = (d > 0.0f) ? rsqrtf(d) : 0.0f;
    }
}

// ---------------------------------------------------------------------------
// 2) h1 = x @ w1   [100000 x 256] @ [256 x 32], f32 WMMA 16x16x4
//    one wave per 16x16 output tile, 4 waves (row tiles) per block
// ---------------------------------------------------------------------------
__global__ __launch_bounds__(128)
void gcn_gemm1_wmma(const float* __restrict__ x,
                    const float* __restrict__ w1,
                    float* __restrict__ h1) {
    const int lane   = threadIdx.x;            // 0..31
    const int rowTile = blockIdx.x * blockDim.y + threadIdx.y;
    const int rowBase = rowTile * 16;
    if (rowBase >= N_NODES) return;            // wave-uniform guard (EXEC all-1 inside)
    const int colBase = blockIdx.y * 16;       // grid.y = HID/16 = 2

    const int m  = lane & 15;                  // matrix row (A) / column (B,D)
    const int kh = (lane >> 4) << 1;           // upper half-wave holds K+2,K+3

    v8f acc = {};
    const float* xrow = x + (size_t)(rowBase + m) * IN_C;

    #pragma unroll 8
    for (int k = 0; k < IN_C; k += 4) {
        v2f a, b;
        a.x = xrow[k + kh];                            // A[m][k+kh]
        a.y = xrow[k + kh + 1];                        // A[m][k+kh+1]
        b.x = w1[(size_t)(k + kh)     * HID + colBase + m];  // B[k+kh][n]
        b.y = w1[(size_t)(k + kh + 1) * HID + colBase + m];  // B[k+kh+1][n]
        acc = __builtin_amdgcn_wmma_f32_16x16x4_f32(
                  false, a, false, b, (short)0, acc, false, false);
    }

    // C/D layout: VGPR v -> row (v + 8*(lane>=16)), col = lane&15
    const int row0 = rowBase + ((lane >> 4) << 3);
    #pragma unroll
    for (int v = 0; v < 8; ++v) {
        h1[(size_t)(row0 + v) * HID + colBase + m] = acc[v];
    }
}

// ---------------------------------------------------------------------------
// 3) layer-1 aggregation: lane = channel (32), one edge per wave row
// ---------------------------------------------------------------------------
__global__ __launch_bounds__(256)
void gcn_agg1_kernel(const long long* __restrict__ src,
                     const long long* __restrict__ dst,
                     const float* __restrict__ dinv,
                     const float* __restrict__ h1,
                     float* __restrict__ agg1) {
    int e = blockIdx.x * blockDim.y + threadIdx.y;
    if (e >= N_EDGES) return;
    long long s = src[e];
    long long d = dst[e];
    float nrm = dinv[s] * dinv[d];
    int c = threadIdx.x;                        // coalesced 128B gather/scatter
    float v = h1[(size_t)s * HID + c] * nrm;
    atomicAdd(&agg1[(size_t)d * HID + c], v);
}

// ---------------------------------------------------------------------------
// 4) h2 = relu(agg1 + b1) @ w2   [100000 x 32] @ [32 x 64], f32 WMMA 16x16x4
//    relu + bias fused into the A-operand load
// ---------------------------------------------------------------------------
__global__ __launch_bounds__(128)
void gcn_gemm2_wmma(const float* __restrict__ agg1,
                    const float* __restrict__ b1,
                    const float* __restrict__ w2,
                    float* __restrict__ h2) {
    const int lane    = threadIdx.x;
    const int rowTile = blockIdx.x * blockDim.y + threadIdx.y;
    const int rowBase = rowTile * 16;
    if (rowBase >= N_NODES) return;
    const int colBase = blockIdx.y * 16;        // grid.y = OUT_C/16 = 4

    const int m  = lane & 15;
    const int kh = (lane >> 4) << 1;

    v8f acc = {};
    const float* arow = agg1 + (size_t)(rowBase + m) * HID;

    #pragma unroll
    for (int k = 0; k < HID; k += 4) {
        v2f a, b;
        a.x = fmaxf(arow[k + kh]     + b1[k + kh],     0.0f);
        a.y = fmaxf(arow[k + kh + 1] + b1[k + kh + 1], 0.0f);
        b.x = w2[(size_t)(k + kh)     * OUT_C + colBase + m];
        b.y = w2[(size_t)(k + kh + 1) * OUT_C + colBase + m];
        acc = __builtin_amdgcn_wmma_f32_16x16x4_f32(
                  false, a, false, b, (short)0, acc, false, false);
    }

    const int row0 = rowBase + ((lane >> 4) << 3);
    #pragma unroll
    for (int v = 0; v < 8; ++v) {
        h2[(size_t)(row0 + v) * OUT_C + colBase + m] = acc[v];
    }
}

// ---------------------------------------------------------------------------
// 5) layer-2 aggregation: lane = channel (64 -> 2 waves per edge)
//    accumulates directly into d_out (pre-zeroed)
// ---------------------------------------------------------------------------
__global__ __launch_bounds__(256)
void gcn_agg2_kernel(const long long* __restrict__ src,
                     const long long* __restrict__ dst,
                     const float* __restrict__ dinv,
                     const float* __restrict__ h2,
                     float* __restrict__ out) {
    int e = blockIdx.x * blockDim.y + threadIdx.y;
    if (e >= N_EDGES) return;
    long long s = src[e];
    long long d = dst[e];
    float nrm = dinv[s] * dinv[d];
    int c = threadIdx.x;                        // 0..63, coalesced 256B
    float v = h2[(size_t)s * OUT_C + c] * nrm;
    atomicAdd(&out[(size_t)d * OUT_C + c], v);
}

// ---------------------------------------------------------------------------
// 6) out = log_softmax(out + b2) ; one wave per node, 2 channels per lane
// ---------------------------------------------------------------------------
__global__ __launch_bounds__(256)
void gcn_logsoftmax_kernel(float* __restrict__ out,
                           const float* __restrict__ b2) {
    int node = blockIdx.x * blockDim.y + threadIdx.y;
    if (node >= N_NODES) return;
    int lane = threadIdx.x;                     // 0..31
    float* row = out + (size_t)node * OUT_C;

    float v0 = row[lane]      + b2[lane];
    float v1 = row[lane + 32] + b2[lane + 32];

    float mx = fmaxf(v0, v1);
    #pragma unroll
    for (int off = 16; off > 0; off >>= 1)
        mx = fmaxf(mx, __shfl_xor(mx, off, 32));

    float s = __expf(v0 - mx) + __expf(v1 - mx);
    #pragma unroll
    for (int off = 16; off > 0; off >>= 1)
        s += __shfl_xor(s, off, 32);

    float lse = __logf(s) + mx;
    row[lane]      = v0 - lse;
    row[lane + 32] = v1 - lse;
}

// ---------------------------------------------------------------------------
extern "C" void kernel_launch(void* const* d_in, const int* in_sizes, int n_in,
                              void* d_out, int out_size, void* d_ws, size_t ws_size,
                              hipStream_t stream) {
    (void)in_sizes; (void)n_in; (void)out_size; (void)ws_size;

    const float*     x   = (const float*)d_in[0];
    const long long* ei  = (const long long*)d_in[1];   // [2, E] int64
    const float*     w1  = (const float*)d_in[2];
    const float*     b1  = (const float*)d_in[3];
    const float*     w2  = (const float*)d_in[4];
    const float*     b2  = (const float*)d_in[5];
    const long long* src = ei;
    const long long* dst = ei + N_EDGES;
    float*           out = (float*)d_out;

    // workspace layout (51.6 MB total)
    float* dinv = (float*)d_ws;                               // N
    float* h1   = dinv + N_NODES;                             // N*32
    float* agg1 = h1   + (size_t)N_NODES * HID;               // N*32
    float* h2   = agg1 + (size_t)N_NODES * HID;               // N*64

    hipMemsetAsync(dinv, 0, (size_t)N_NODES * sizeof(float), stream);
    hipMemsetAsync(agg1, 0, (size_t)N_NODES * HID * sizeof(float), stream);
    hipMemsetAsync(out,  0, (size_t)N_NODES * OUT_C * sizeof(float), stream);

    gcn_deg_kernel<<<(N_EDGES + 255) / 256, 256, 0, stream>>>(dst, dinv);
    gcn_dinv_kernel<<<(N_NODES + 255) / 256, 256, 0, stream>>>(dinv);

    {   // GEMM1: 6250 row tiles, 2 col tiles
        dim3 blk(32, 4);
        dim3 grd((N_NODES / 16 + 3) / 4, HID / 16);
        gcn_gemm1_wmma<<<grd, blk, 0, stream>>>(x, w1, h1);
    }
    {   // layer-1 scatter
        dim3 blk(32, 8);
        dim3 grd((N_EDGES + 7) / 8);
        gcn_agg1_kernel<<<grd, blk, 0, stream>>>(src, dst, dinv, h1, agg1);
    }
    {   // GEMM2: 6250 row tiles, 4 col tiles (relu+bias fused)
        dim3 blk(32, 4);
        dim3 grd((N_NODES / 16 + 3) / 4, OUT_C / 16);
        gcn_gemm2_wmma<<<grd, blk, 0, stream>>>(agg1, b1, w2, h2);
    }
    {   // layer-2 scatter into d_out
        dim3 blk(64, 4);
        dim3 grd((N_EDGES + 3) / 4);
        gcn_agg2_kernel<<<grd, blk, 0, stream>>>(src, dst, dinv, h2, out);
    }
    {   // bias + log_softmax
        dim3 blk(32, 8);
        dim3 grd((N_NODES + 7) / 8);
        gcn_logsoftmax_kernel<<<grd, blk, 0, stream>>>(out, b2);
    }
}